// GCNv2_18786186952918
// MI455X (gfx1250) — compile-verified
//
#include <hip/hip_runtime.h>
#include <hip/hip_bf16.h>

// GCN forward for MI455X (gfx1250).
// GEMMs via V_WMMA_F32_16X16X4_F32 (fp32-exact, matches reference numerics).
// Edge scatter via global_atomic_add_f32; feature matrix (51.2MB) is L2-resident.
//
// Workspace layout (needs ~161 MB):
//   bufA (tmp = h@W), bufB, bufC : N*128 f32 each (ping-pong h/accum)
//   deg, dinv : N f32 ; norm : E f32 ; pooled : 64*128 f32 ; cnt : 64 f32

#define GCN_N 100000
#define GCN_E 1600000
#define GCN_G 64
#define GCN_D 128
#define GCN_DOUT 40
#define GCN_LAYERS 4

typedef __attribute__((ext_vector_type(2))) float v2f;
typedef __attribute__((ext_vector_type(8))) float v8f;

#define LDSTRIDE 132   // pad 128-float rows to avoid 16-way LDS bank conflicts

// ---------------- WMMA GEMM: out[N x 128] = H[N x 128] @ W[128 x 128] -------
// One block = 16-row strip of H (staged in LDS), 8 waves = 8 column tiles.
__global__ void __launch_bounds__(256)
gcn_gemm_wmma(const float* __restrict__ H, const float* __restrict__ W,
              float* __restrict__ out)
{
    __shared__ float Atile[16 * LDSTRIDE];
    const int t = threadIdx.x;
    const int blockRow = blockIdx.x;           // which 16-row strip

    // Stage A: 16x128 f32 (8KB) cooperatively, two float4 per thread.
    {
        const int row  = t >> 4;               // 0..15
        const int col0 = (t & 15) * 8;         // 0,8,...,120
        const float4* g =
            (const float4*)(H + ((size_t)blockRow * 16 + row) * GCN_D + col0);
        float4 a0 = g[0];
        float4 a1 = g[1];
        float4* l = (float4*)(Atile + row * LDSTRIDE + col0);
        l[0] = a0;
        l[1] = a1;
    }
    __syncthreads();

    const int wave = t >> 5;                   // 0..7 -> output column tile
    const int lane = t & 31;
    const int m    = lane & 15;                // row within tile (A/M, B/N)
    const int koff = (lane >> 4) << 1;         // 0 (lanes 0-15) or 2 (16-31)
    const int col  = wave * 16 + m;            // global output column

    v8f c = {};
    #pragma unroll 4
    for (int k0 = 0; k0 < GCN_D; k0 += 4) {
        // A fragment 16x4: lane holds A[m, k0+koff] and A[m, k0+koff+1]
        const float* ap = Atile + m * LDSTRIDE + k0 + koff;
        v2f a; a.x = ap[0]; a.y = ap[1];
        // B fragment 4x16: lane holds W[k0+koff, col] and W[k0+koff+1, col]
        const float* bp = W + (size_t)(k0 + koff) * GCN_D + col;
        v2f b; b.x = bp[0]; b.y = bp[GCN_D];
        c = __builtin_amdgcn_wmma_f32_16x16x4_f32(
                false, a, false, b, (short)0, c, false, false);
    }

    // D 16x16 f32: VGPR g -> row (g + 8*(lane>>4)), col = lane&15
    const int rowBase = blockRow * 16 + ((lane >> 4) << 3);
    float* o = out + (size_t)rowBase * GCN_D + col;
    #pragma unroll
    for (int g = 0; g < 8; ++g)
        o[(size_t)g * GCN_D] = c[g];
}

// ---------------- elementwise / graph kernels --------------------------------
__global__ void gcn_fill_f32(float* __restrict__ p, float v, int n)
{
    int i = blockIdx.x * blockDim.x + threadIdx.x;
    if (i < n) p[i] = v;
}

__global__ void gcn_degree(const int* __restrict__ dst, float* __restrict__ deg)
{
    int e = blockIdx.x * blockDim.x + threadIdx.x;
    if (e < GCN_E) atomicAdd(&deg[dst[e]], 1.0f);
}

__global__ void gcn_rsqrt(const float* __restrict__ deg, float* __restrict__ dinv)
{
    int i = blockIdx.x * blockDim.x + threadIdx.x;
    if (i < GCN_N) dinv[i] = rsqrtf(deg[i]);   // deg >= 1 (self-loop folded in)
}

__global__ void gcn_norm(const int* __restrict__ src, const int* __restrict__ dst,
                         const float* __restrict__ dinv, float* __restrict__ nrm)
{
    int e = blockIdx.x * blockDim.x + threadIdx.x;
    if (e < GCN_E) nrm[e] = dinv[src[e]] * dinv[dst[e]];
}

// One wave32 per edge; each lane moves a float4 (32 lanes * 4 = 128 feats).
__global__ void __launch_bounds__(256)
gcn_edge_scatter(const int* __restrict__ src, const int* __restrict__ dst,
                 const float* __restrict__ nrm, const float* __restrict__ tmp,
                 float* __restrict__ accum)
{
    int e = blockIdx.x * (blockDim.x >> 5) + (threadIdx.x >> 5);
    if (e >= GCN_E) return;
    const int lane = threadIdx.x & 31;
    const int s = src[e];
    const int d = dst[e];
    const float w = nrm[e];
    float4 v = ((const float4*)(tmp + (size_t)s * GCN_D))[lane];
    float* o = accum + (size_t)d * GCN_D + lane * 4;
    atomicAdd(o + 0, v.x * w);
    atomicAdd(o + 1, v.y * w);
    atomicAdd(o + 2, v.z * w);
    atomicAdd(o + 3, v.w * w);
}

// h_out = relu(accum + tmp * dinv^2 (self-loop) + bias); in-place on accum ok.
__global__ void gcn_finish(const float* __restrict__ accum,
                           const float* __restrict__ tmp,
                           const float* __restrict__ dinv,
                           const float* __restrict__ bias,
                           float* __restrict__ hout)
{
    int idx = blockIdx.x * blockDim.x + threadIdx.x;
    if (idx >= GCN_N * GCN_D) return;
    int i = idx >> 7;
    int f = idx & 127;
    float s = dinv[i];
    float v = accum[idx] + tmp[idx] * (s * s) + bias[f];
    hout[idx] = fmaxf(v, 0.0f);
}

__global__ void gcn_pool_sum(const float* __restrict__ h,
                             const int* __restrict__ batch,
                             float* __restrict__ pooled)
{
    int idx = blockIdx.x * blockDim.x + threadIdx.x;
    if (idx >= GCN_N * GCN_D) return;
    int i = idx >> 7;
    int f = idx & 127;
    atomicAdd(&pooled[batch[i] * GCN_D + f], h[idx]);
}

__global__ void gcn_pool_cnt(const int* __restrict__ batch, float* __restrict__ cnt)
{
    int i = blockIdx.x * blockDim.x + threadIdx.x;
    if (i < GCN_N) atomicAdd(&cnt[batch[i]], 1.0f);
}

__global__ void gcn_final(const float* __restrict__ pooled,
                          const float* __restrict__ cnt,
                          const float* __restrict__ Wlin,
                          const float* __restrict__ blin,
                          float* __restrict__ out)
{
    int idx = blockIdx.x * blockDim.x + threadIdx.x;
    if (idx >= GCN_G * GCN_DOUT) return;
    int g = idx / GCN_DOUT;
    int o = idx - g * GCN_DOUT;
    float inv = 1.0f / fmaxf(cnt[g], 1.0f);
    float acc = blin[o];
    #pragma unroll 4
    for (int f = 0; f < GCN_D; ++f)
        acc += pooled[g * GCN_D + f] * inv * Wlin[f * GCN_DOUT + o];
    out[idx] = acc;
}

// ---------------- host-side orchestration ------------------------------------
static inline size_t alignUp(size_t x) { return (x + 255) & ~(size_t)255; }

extern "C" void kernel_launch(void* const* d_in, const int* in_sizes, int n_in,
                              void* d_out, int out_size, void* d_ws, size_t ws_size,
                              hipStream_t stream) {
    (void)in_sizes; (void)n_in; (void)out_size; (void)ws_size;

    const float* x     = (const float*)d_in[0];
    const int*   ei    = (const int*)  d_in[1];   // [2, E] flat
    const int*   batch = (const int*)  d_in[2];
    const float* W_in  = (const float*)d_in[3];
    const float* b_in  = (const float*)d_in[4];
    const float* W_h   = (const float*)d_in[5];   // [3,128,128]
    const float* b_h   = (const float*)d_in[6];   // [3,128]
    const float* W_lin = (const float*)d_in[7];
    const float* b_lin = (const float*)d_in[8];
    float* out = (float*)d_out;

    const int* src = ei;
    const int* dst = ei + GCN_E;

    // workspace carve-up
    char* ws = (char*)d_ws;
    size_t off = 0;
    const size_t featBytes = (size_t)GCN_N * GCN_D * sizeof(float);
    float* bufA = (float*)(ws + off); off = alignUp(off + featBytes);  // tmp = h@W
    float* bufB = (float*)(ws + off); off = alignUp(off + featBytes);
    float* bufC = (float*)(ws + off); off = alignUp(off + featBytes);
    float* deg  = (float*)(ws + off); off = alignUp(off + (size_t)GCN_N * 4);
    float* dinv = (float*)(ws + off); off = alignUp(off + (size_t)GCN_N * 4);
    float* nrm  = (float*)(ws + off); off = alignUp(off + (size_t)GCN_E * 4);
    float* pooled = (float*)(ws + off); off = alignUp(off + (size_t)GCN_G * GCN_D * 4);
    float* cnt  = (float*)(ws + off); off = alignUp(off + (size_t)GCN_G * 4);

    const int T = 256;
    const int gN   = (GCN_N + T - 1) / T;
    const int gE   = (GCN_E + T - 1) / T;
    const int gND  = (GCN_N * GCN_D + T - 1) / T;
    const int gTile = GCN_N / 16;                 // 6250 (exact)
    const int gEdgeWave = (GCN_E + 7) / 8;        // 8 waves/block, 1 edge/wave

    // ---- degree / norm (self-loops: deg starts at 1, self term in epilogue)
    gcn_fill_f32<<<gN, T, 0, stream>>>(deg, 1.0f, GCN_N);
    gcn_degree  <<<gE, T, 0, stream>>>(dst, deg);
    gcn_rsqrt   <<<gN, T, 0, stream>>>(deg, dinv);
    gcn_norm    <<<gE, T, 0, stream>>>(src, dst, dinv, nrm);

    // ---- layers (ping-pong: h goes x -> bufB -> bufC -> bufB -> bufC)
    const float* h_in = x;
    float* accums[GCN_LAYERS] = { bufB, bufC, bufB, bufC };
    for (int l = 0; l < GCN_LAYERS; ++l) {
        const float* Wl = (l == 0) ? W_in : (W_h + (size_t)(l - 1) * GCN_D * GCN_D);
        const float* bl = (l == 0) ? b_in : (b_h + (size_t)(l - 1) * GCN_D);
        float* accum = accums[l];

        gcn_gemm_wmma<<<gTile, T, 0, stream>>>(h_in, Wl, bufA);
        gcn_fill_f32 <<<gND, T, 0, stream>>>(accum, 0.0f, GCN_N * GCN_D);
        gcn_edge_scatter<<<gEdgeWave, T, 0, stream>>>(src, dst, nrm, bufA, accum);
        gcn_finish   <<<gND, T, 0, stream>>>(accum, bufA, dinv, bl, accum);
        h_in = accum;
    }

    // ---- global mean pool + final linear
    gcn_fill_f32<<<(GCN_G * GCN_D + T - 1) / T, T, 0, stream>>>(pooled, 0.0f, GCN_G * GCN_D);
    gcn_fill_f32<<<1, T, 0, stream>>>(cnt, 0.0f, GCN_G);
    gcn_pool_sum<<<gND, T, 0, stream>>>(h_in, batch, pooled);
    gcn_pool_cnt<<<gN, T, 0, stream>>>(batch, cnt);
    gcn_final<<<(GCN_G * GCN_DOUT + T - 1) / T, T, 0, stream>>>(pooled, cnt, W_lin, b_lin, out);
}